// SVDpp_63831803953698
// MI455X (gfx1250) — compile-verified
//
#include <hip/hip_runtime.h>

typedef __attribute__((ext_vector_type(2))) float v2f;
typedef __attribute__((ext_vector_type(8))) float v8f;

#define EMB    64
#define WAVES  16              // samples per block (one wave32 per sample)
#define BLOCK  (WAVES * 32)

// lower_bound over sorted int array (uniform across the wave)
static __device__ __forceinline__ int lbound(const int* __restrict__ seg, int n, int key) {
    int lo = 0, hi = n;
    while (lo < hi) {
        int mid = (lo + hi) >> 1;
        if (seg[mid] < key) lo = mid + 1; else hi = mid;
    }
    return lo;
}

// Pool rows [s,e) of `table` (gathered via idx), each lane accumulating dims
// [2*lane, 2*lane+1]. Unroll-4 to keep >=4 gathers in flight per wave.
static __device__ __forceinline__ v2f pool_rows(const float* __restrict__ table,
                                                const int* __restrict__ idx,
                                                int s, int e, int lane) {
    v2f a0 = {0.f, 0.f}, a1 = {0.f, 0.f}, a2 = {0.f, 0.f}, a3 = {0.f, 0.f};
    int i = s;
    for (; i + 3 < e; i += 4) {
        int j0 = idx[i], j1 = idx[i + 1], j2 = idx[i + 2], j3 = idx[i + 3];
        a0 += *(const v2f*)(table + (size_t)j0 * EMB + 2 * lane);
        a1 += *(const v2f*)(table + (size_t)j1 * EMB + 2 * lane);
        a2 += *(const v2f*)(table + (size_t)j2 * EMB + 2 * lane);
        a3 += *(const v2f*)(table + (size_t)j3 * EMB + 2 * lane);
    }
    for (; i < e; ++i) {
        int j0 = idx[i];
        a0 += *(const v2f*)(table + (size_t)j0 * EMB + 2 * lane);
    }
    v2f acc = (a0 + a1) + (a2 + a3);
    float cnt = (float)(e - s);
    float scale = (cnt > 0.f) ? rsqrtf(cnt) : 0.f;
    acc.x *= scale; acc.y *= scale;
    return acc;
}

__global__ __launch_bounds__(BLOCK) void svdpp_kernel(
    const int*   __restrict__ sci_ids,  const int*   __restrict__ pap_ids,
    const int*   __restrict__ p_idx,    const int*   __restrict__ p_seg,
    const int*   __restrict__ w_idx,    const int*   __restrict__ w_seg,
    const float* __restrict__ sfac,     const float* __restrict__ pfac,
    const float* __restrict__ sbias,    const float* __restrict__ pbias,
    const float* __restrict__ ifac,     const float* __restrict__ iwl,
    const float* __restrict__ gbias,
    float*       __restrict__ out,
    int TPn, int TWn, int Bn)
{
    __shared__ float lds_comb[WAVES][EMB];   // A: pooled+user vectors, row = sample-in-block
    __shared__ float lds_p[WAVES][EMB];      // B^T: paper embeddings, row = sample-in-block
    __shared__ float lds_bias[WAVES];

    const int wave = threadIdx.x >> 5;
    const int lane = threadIdx.x & 31;
    const int b    = blockIdx.x * WAVES + wave;

    // -------- pooling phase: one wave per sample --------
    if (b < Bn) {
        const int sid = sci_ids[b];
        const int pid = pap_ids[b];

        v2f semb = *(const v2f*)(sfac + (size_t)sid * EMB + 2 * lane);
        v2f pemb = *(const v2f*)(pfac + (size_t)pid * EMB + 2 * lane);

        int ps = lbound(p_seg, TPn, b), pe = lbound(p_seg, TPn, b + 1);
        int wsb = lbound(w_seg, TWn, b), web = lbound(w_seg, TWn, b + 1);

        v2f yu  = pool_rows(ifac, p_idx, ps,  pe,  lane);
        v2f ywl = pool_rows(iwl,  w_idx, wsb, web, lane);

        v2f comb = semb + yu + ywl;

        *(v2f*)&lds_comb[wave][2 * lane] = comb;
        *(v2f*)&lds_p[wave][2 * lane]    = pemb;
        if (lane == 0)
            lds_bias[wave] = sbias[sid] + pbias[pid] + gbias[0];
    } else {
        // pad block: zero contribution so the WMMA tile is harmless
        v2f z = {0.f, 0.f};
        *(v2f*)&lds_comb[wave][2 * lane] = z;
        *(v2f*)&lds_p[wave][2 * lane]    = z;
        if (lane == 0) lds_bias[wave] = 0.f;
    }

    __syncthreads();

    // -------- WMMA phase: wave 0 computes 16 dots as diag(A x B) --------
    // D[m][n] = sum_k comb[m][k] * pemb[n][k]; we keep only m == n.
    if (wave == 0) {
        const int half = lane >> 4;     // 0: lanes 0-15, 1: lanes 16-31
        const int ln   = lane & 15;

        v8f c = {};
        #pragma unroll
        for (int k0 = 0; k0 < EMB; k0 += 4) {
            // 32-bit A 16x4 layout: VGPR0 = K+0 (lanes 0-15) / K+2 (lanes 16-31),
            // VGPR1 = K+1 / K+3. B (4x16) mirrors it with N striped across lanes.
            const int k = k0 + 2 * half;
            v2f a, bm;
            a.x  = lds_comb[ln][k];  a.y  = lds_comb[ln][k + 1];
            bm.x = lds_p[ln][k];     bm.y = lds_p[ln][k + 1];
            c = __builtin_amdgcn_wmma_f32_16x16x4_f32(
                    /*neg_a=*/false, a, /*neg_b=*/false, bm,
                    /*c_mod=*/(short)0, c, /*reuse_a=*/false, /*reuse_b=*/false);
        }

        // Diagonal extraction from C/D layout:
        //   M=0..7  -> lane M      , VGPR M
        //   M=8..15 -> lane M + 16 , VGPR M - 8
        int m = -1, r = 0;
        if (lane < 8)        { m = lane;      r = lane;      }
        else if (lane >= 24) { m = lane - 16; r = lane - 24; }

        float v = c[0];
        #pragma unroll
        for (int j = 1; j < 8; ++j) v = (r == j) ? c[j] : v;

        if (m >= 0) {
            int ob = blockIdx.x * WAVES + m;
            if (ob < Bn) out[ob] = v + lds_bias[m];
        }
    }
}

extern "C" void kernel_launch(void* const* d_in, const int* in_sizes, int n_in,
                              void* d_out, int out_size, void* d_ws, size_t ws_size,
                              hipStream_t stream) {
    const int*   sci_ids = (const int*)d_in[0];
    const int*   pap_ids = (const int*)d_in[1];
    const int*   p_idx   = (const int*)d_in[2];
    const int*   p_seg   = (const int*)d_in[3];
    const int*   w_idx   = (const int*)d_in[4];
    const int*   w_seg   = (const int*)d_in[5];
    const float* sfac    = (const float*)d_in[6];
    const float* pfac    = (const float*)d_in[7];
    const float* sbias   = (const float*)d_in[8];
    const float* pbias   = (const float*)d_in[9];
    const float* ifac    = (const float*)d_in[10];
    const float* iwl     = (const float*)d_in[11];
    const float* gbias   = (const float*)d_in[12];
    float*       out     = (float*)d_out;

    const int Bn  = in_sizes[0];
    const int TPn = in_sizes[2];
    const int TWn = in_sizes[4];

    const int nblk = (Bn + WAVES - 1) / WAVES;
    svdpp_kernel<<<nblk, BLOCK, 0, stream>>>(
        sci_ids, pap_ids, p_idx, p_seg, w_idx, w_seg,
        sfac, pfac, sbias, pbias, ifac, iwl, gbias,
        out, TPn, TWn, Bn);
}